// ChaCha20Cipher_16209206575200
// MI455X (gfx1250) — compile-verified
//
#include <hip/hip_runtime.h>
#include <cstddef>

// Soft-ChaCha20 on gfx1250, v4 (final form):
//  - normalized domain (u = x / 2^32): exact power-of-2 rescale of reference
//  - two rows per thread as float2 -> V_PK_*_F32 packed math
//  - soft_xor sigmoids via gfx1250 v_tanh_f32 (TRANS, 1 op per sigmoid)
//  - soft_add "sigmoid" is bit-exactly a {0,0.5,1} step at f32 granularity:
//    wrap = fma(0.5, med3(2^25*(su-1), -1, 1), 0.5); out = su - wrap
//    (single subtraction — exact scaled mirror of reference s - M*wrap)
//  - fract via v_fract_f32 (== x - floor(x), matches jnp.mod(.,1))

typedef float v2f __attribute__((ext_vector_type(2)));

namespace {

constexpr float MODF  = 4294967296.0f;              // 2^32
constexpr float INV_M = 2.32830643653869629e-10f;   // 2^-32 (exact)
constexpr float BIG   = 33554432.0f;                // 2^25: min nonzero |su-1| is
                                                    // 2^-24 -> |BIG*(su-1)| >= 2

__device__ __forceinline__ float htanh(float x) {
#if __has_builtin(__builtin_amdgcn_tanhf)
  return __builtin_amdgcn_tanhf(x);                 // v_tanh_f32
#else
  float r;
  asm("v_tanh_f32 %0, %1" : "=v"(r) : "v"(x));
  return r;
#endif
}
__device__ __forceinline__ float hfract(float x) {
#if __has_builtin(__builtin_amdgcn_fractf)
  return __builtin_amdgcn_fractf(x);                // v_fract_f32
#else
  return x - __builtin_floorf(x);
#endif
}

__device__ __forceinline__ v2f vtanh(v2f a) {
  v2f r;
  r.x = htanh(a.x);
  r.y = htanh(a.y);
  return r;
}
__device__ __forceinline__ v2f vfract(v2f a) {
  v2f r;
  r.x = hfract(a.x);
  r.y = hfract(a.y);
  return r;
}
__device__ __forceinline__ v2f vsignsat(v2f a) {   // med3(x,-1,1): exact sign-step
  v2f r;
  r.x = __builtin_amdgcn_fmed3f(a.x, -1.0f, 1.0f);
  r.y = __builtin_amdgcn_fmed3f(a.y, -1.0f, 1.0f);
  return r;
}
__device__ __forceinline__ v2f vfma(v2f a, v2f b, v2f c) {
  return __builtin_elementwise_fma(a, b, c);
}
__device__ __forceinline__ v2f splat(float s) {
  v2f r = {s, s};
  return r;
}

// soft_add: su = ux+uy; reference wrap = sigmoid(5M(su-1)), which at f32
// granularity (arg is 0 or |arg| >= 1280) is exactly {0, 0.5, 1}.
// t = med3(2^25*(su-1), -1, 1) in {-1,0,1}; wrap = 0.5 + 0.5*t (exact);
// out = su - wrap as ONE subtraction (preserves tiny su exactly, like ref).
__device__ __forceinline__ v2f sadd(v2f x, v2f y) {
  v2f s = x + y;
  v2f t = vsignsat(vfma(splat(BIG), s, splat(-BIG)));
  v2f wrap = vfma(splat(0.5f), t, splat(0.5f));
  return s - wrap;
}

// soft_xor: tx = tanh(2.5*ux - 1.25) (smooth region, genuine tanh);
// out_u = 0.5*(1 - tx*ty)
__device__ __forceinline__ v2f sxor(v2f x, v2f y) {
  v2f tx = vtanh(vfma(splat(2.5f), x, splat(-1.25f)));
  v2f ty = vtanh(vfma(splat(2.5f), y, splat(-1.25f)));
  return vfma(splat(-0.5f), tx * ty, splat(0.5f));
}

// smooth_rotate (input u in [0,1]: always follows soft_xor):
// out_u = fract(fract(u*2^R) + u*2^(R-32)); right term never wraps.
template <int R>
__device__ __forceinline__ v2f srot(v2f u) {
  constexpr float P  = (float)(1u << R);
  constexpr float PN = 1.0f / (float)(1ull << (32 - R));
  v2f l = vfract(u * splat(P));
  return vfract(vfma(u, splat(PN), l));
}

__device__ __forceinline__ void qround(v2f& a, v2f& b, v2f& c, v2f& d) {
  a = sadd(a, b); d = sxor(d, a); d = srot<16>(d);
  c = sadd(c, d); b = sxor(b, c); b = srot<12>(b);
  a = sadd(a, b); d = sxor(d, a); d = srot<8>(d);
  c = sadd(c, d); b = sxor(b, c); b = srot<7>(b);
}

} // namespace

__global__ __launch_bounds__(256) void chacha_soft4_kernel(
    const float* __restrict__ in, float* __restrict__ out, int nrows) {
  int pair = blockIdx.x * blockDim.x + threadIdx.x;
  int i0 = pair * 2;
  if (i0 >= nrows) return;
  int i1 = (i0 + 1 < nrows) ? (i0 + 1) : i0;

  const float4* qa = reinterpret_cast<const float4*>(in + (size_t)i0 * 16);
  const float4* qb = reinterpret_cast<const float4*>(in + (size_t)i1 * 16);
  float4 A0 = qa[0], A1 = qa[1], A2 = qa[2], A3 = qa[3];
  float4 B0 = qb[0], B1 = qb[1], B2 = qb[2], B3 = qb[3];

  float ra[16] = {A0.x, A0.y, A0.z, A0.w, A1.x, A1.y, A1.z, A1.w,
                  A2.x, A2.y, A2.z, A2.w, A3.x, A3.y, A3.z, A3.w};
  float rb[16] = {B0.x, B0.y, B0.z, B0.w, B1.x, B1.y, B1.z, B1.w,
                  B2.x, B2.y, B2.z, B2.w, B3.x, B3.y, B3.z, B3.w};

  v2f w[16], s0[16];
#pragma unroll
  for (int i = 0; i < 16; ++i) {
    v2f v = {ra[i], rb[i]};
    w[i] = v * splat(INV_M);   // exact normalization to u-domain
    s0[i] = w[i];
  }

  // Rolled double-round loop keeps the hot body I$-resident; 8 unrolled
  // quarter-rounds x 2 packed rows provide ILP so v_tanh_f32 (TRANS)
  // co-executes with the packed-FMA stream.
#pragma unroll 1
  for (int r = 0; r < 10; ++r) {
    qround(w[0], w[4], w[8],  w[12]);
    qround(w[1], w[5], w[9],  w[13]);
    qround(w[2], w[6], w[10], w[14]);
    qround(w[3], w[7], w[11], w[15]);
    qround(w[0], w[5], w[10], w[15]);
    qround(w[1], w[6], w[11], w[12]);
    qround(w[2], w[7], w[8],  w[13]);
    qround(w[3], w[4], w[9],  w[14]);
  }

  // out = mod(w + state, 2^32) -> fract(uw + us0) * 2^32 (exact rescale)
  float oa[16], ob[16];
#pragma unroll
  for (int i = 0; i < 16; ++i) {
    v2f v = vfract(w[i] + s0[i]) * splat(MODF);
    oa[i] = v.x;
    ob[i] = v.y;
  }

  float4* za = reinterpret_cast<float4*>(out + (size_t)i0 * 16);
  za[0] = make_float4(oa[0],  oa[1],  oa[2],  oa[3]);
  za[1] = make_float4(oa[4],  oa[5],  oa[6],  oa[7]);
  za[2] = make_float4(oa[8],  oa[9],  oa[10], oa[11]);
  za[3] = make_float4(oa[12], oa[13], oa[14], oa[15]);
  if (i0 + 1 < nrows) {
    float4* zb = reinterpret_cast<float4*>(out + (size_t)i1 * 16);
    zb[0] = make_float4(ob[0],  ob[1],  ob[2],  ob[3]);
    zb[1] = make_float4(ob[4],  ob[5],  ob[6],  ob[7]);
    zb[2] = make_float4(ob[8],  ob[9],  ob[10], ob[11]);
    zb[3] = make_float4(ob[12], ob[13], ob[14], ob[15]);
  }
}

extern "C" void kernel_launch(void* const* d_in, const int* in_sizes, int n_in,
                              void* d_out, int out_size, void* d_ws, size_t ws_size,
                              hipStream_t stream) {
  const float* state = (const float*)d_in[0];
  float* out = (float*)d_out;
  int nrows = in_sizes[0] / 16;          // 4,194,304 rows of 16 words
  int pairs = (nrows + 1) / 2;           // two rows per thread
  const int threads = 256;
  int blocks = (pairs + threads - 1) / threads;
  chacha_soft4_kernel<<<blocks, threads, 0, stream>>>(state, out, nrows);
}